// STinSTMulti_90314572300622
// MI455X (gfx1250) — compile-verified
//
#include <hip/hip_runtime.h>
#include <hip/hip_bf16.h>

typedef float v2f __attribute__((ext_vector_type(2)));
typedef float v8f __attribute__((ext_vector_type(8)));

#define B_  256
#define C_  32
#define H_  64
#define W_  64
#define K1  (C_*H_*W_)      // 131072
#define N1  32
#define KCHUNKS 256
#define KLEN (K1/KCHUNKS)   // 512

// -------- Kernel A: split-K GEMM  partial[kc] = x[m,:]·w1[n,:]  via V_WMMA_F32_16X16X4_F32
// One wave = one (16-row m-tile, all 32 cols) x one K-chunk of 512.
// A fragment (16x4 f32): lane<16 holds (M=lane, K=k0..k0+1); lane>=16 holds (M=lane-16, K=k0+2..k0+3)
// B fragment (4x16 f32): lane<16 holds (N=lane, K=k0..k0+1); lane>=16 holds (N=lane-16, K=k0+2..k0+3)
__global__ void stn_gemm1_wmma(const float* __restrict__ x,
                               const float* __restrict__ w1,
                               float* __restrict__ partials) {
    const int lane = threadIdx.x & 31;
    const int gw   = blockIdx.x * (blockDim.x >> 5) + (threadIdx.x >> 5);
    const int mt   = gw & 15;          // 16 m-tiles
    const int kc   = gw >> 4;          // 256 K-chunks
    const int m0   = mt * 16;
    const int kbase = kc * KLEN;

    const int ln   = lane & 15;
    const int koff = (lane >> 4) * 2;

    const float* aptr  = x  + (size_t)(m0 + ln) * K1 + kbase + koff;
    const float* b0ptr = w1 + (size_t)(ln)      * K1 + kbase + koff;   // cols 0..15
    const float* b1ptr = w1 + (size_t)(ln + 16) * K1 + kbase + koff;   // cols 16..31

    v8f c0 = {};
    v8f c1 = {};
#pragma unroll 4
    for (int k = 0; k < KLEN; k += 4) {
        v2f a  = *(const v2f*)(aptr  + k);
        v2f b0 = *(const v2f*)(b0ptr + k);
        v2f b1 = *(const v2f*)(b1ptr + k);
        c0 = __builtin_amdgcn_wmma_f32_16x16x4_f32(false, a, false, b0, (short)0, c0, false, false);
        c1 = __builtin_amdgcn_wmma_f32_16x16x4_f32(false, a, false, b1, (short)0, c1, false, false);
    }

    // C/D layout: VGPR v, lanes 0-15 -> (M=v, N=lane); lanes 16-31 -> (M=8+v, N=lane-16)
    float* pout = partials + ((size_t)kc * B_ + m0) * N1;
    const int mb = (lane >> 4) * 8;
#pragma unroll
    for (int v = 0; v < 8; ++v) {
        pout[(mb + v) * N1 + ln]      = c0[v];
        pout[(mb + v) * N1 + ln + 16] = c1[v];
    }
}

// -------- Kernel B1: reduce split-K partials, add bias, ReLU -> h1[256,32]
__global__ void stn_reduce_relu(const float* __restrict__ partials,
                                const float* __restrict__ b1v,
                                float* __restrict__ h1) {
    const int tid = blockIdx.x * blockDim.x + threadIdx.x;   // 0..8191 (= m*32+n)
    float s = b1v[tid & (N1 - 1)];
    for (int kc = 0; kc < KCHUNKS; ++kc)
        s += partials[(size_t)kc * (B_ * N1) + tid];         // coalesced across tid
    h1[tid] = fmaxf(s, 0.0f);
}

// -------- Kernel B2: tiny MLP tail: h2 = relu(h1@w2^T+b2); theta = h2@w3^T+b3  (1 block, 256 threads)
__global__ void stn_mlp_tail(const float* __restrict__ h1,
                             const float* __restrict__ w2, const float* __restrict__ b2,
                             const float* __restrict__ w3, const float* __restrict__ b3,
                             float* __restrict__ theta) {
    __shared__ float sw2[32 * 32];
    __shared__ float sw3[6 * 32];
    __shared__ float sb2[32];
    __shared__ float sb3[8];
    const int t = threadIdx.x;
    for (int i = t; i < 1024; i += 256) sw2[i] = w2[i];
    for (int i = t; i < 192;  i += 256) sw3[i] = w3[i];
    if (t < 32) sb2[t] = b2[t];
    if (t < 6)  sb3[t] = b3[t];
    __syncthreads();

    float a[32];
#pragma unroll
    for (int i = 0; i < 32; ++i) a[i] = h1[t * 32 + i];

    float h2[32];
#pragma unroll
    for (int j = 0; j < 32; ++j) {
        float s = sb2[j];
#pragma unroll
        for (int i = 0; i < 32; ++i) s += a[i] * sw2[j * 32 + i];
        h2[j] = fmaxf(s, 0.0f);
    }
#pragma unroll
    for (int j = 0; j < 6; ++j) {
        float s = sb3[j];
#pragma unroll
        for (int i = 0; i < 32; ++i) s += h2[i] * sw3[j * 32 + i];
        theta[t * 6 + j] = s;
    }
}

// -------- Kernel C: affine grid + bilinear sample (zero padding, align_corners=False)
// One thread per output element; w fastest -> near-coalesced gathers for near-identity theta.
__global__ void stn_sample(const float* __restrict__ x,
                           const float* __restrict__ theta,
                           float* __restrict__ out) {
    const int tid = blockIdx.x * blockDim.x + threadIdx.x;
    const int w = tid & 63;
    const int h = (tid >> 6) & 63;
    const int c = (tid >> 12) & 31;
    const int b = tid >> 17;

    const float* th = theta + b * 6;   // uniform within a block -> broadcast via cache
    const float xn = (2.0f * w + 1.0f) * (1.0f / W_) - 1.0f;
    const float yn = (2.0f * h + 1.0f) * (1.0f / H_) - 1.0f;
    const float gxn = th[0] * xn + th[1] * yn + th[2];
    const float gyn = th[3] * xn + th[4] * yn + th[5];
    const float gx = (gxn + 1.0f) * (W_ * 0.5f) - 0.5f;
    const float gy = (gyn + 1.0f) * (H_ * 0.5f) - 0.5f;

    const float x0 = floorf(gx), y0 = floorf(gy);
    const float x1 = x0 + 1.0f,  y1 = y0 + 1.0f;
    const float wx1 = gx - x0, wx0 = 1.0f - wx1;
    const float wy1 = gy - y0, wy0 = 1.0f - wy1;

    const float* img = x + ((size_t)b * C_ + c) * (H_ * W_);
    float acc = 0.0f;
    auto corner = [&](float xf, float yf, float wt) {
        const bool valid = (xf >= 0.0f) & (xf <= (float)(W_ - 1)) &
                           (yf >= 0.0f) & (yf <= (float)(H_ - 1));
        const int xi = (int)fminf(fmaxf(xf, 0.0f), (float)(W_ - 1));
        const int yi = (int)fminf(fmaxf(yf, 0.0f), (float)(H_ - 1));
        const float v = img[yi * W_ + xi];
        acc += valid ? v * wt : 0.0f;
    };
    corner(x0, y0, wx0 * wy0);
    corner(x1, y0, wx1 * wy0);
    corner(x0, y1, wx0 * wy1);
    corner(x1, y1, wx1 * wy1);

    out[tid] = acc;
}

extern "C" void kernel_launch(void* const* d_in, const int* in_sizes, int n_in,
                              void* d_out, int out_size, void* d_ws, size_t ws_size,
                              hipStream_t stream) {
    const float* x   = (const float*)d_in[0];
    const float* w1  = (const float*)d_in[1];
    const float* b1v = (const float*)d_in[2];
    const float* w2  = (const float*)d_in[3];
    const float* b2v = (const float*)d_in[4];
    const float* w3  = (const float*)d_in[5];
    const float* b3v = (const float*)d_in[6];
    float* out = (float*)d_out;

    // Workspace carve-up (all fully overwritten each call -> no init needed)
    float* partials = (float*)d_ws;                          // 256*256*32 = 2M floats (8 MB)
    float* h1    = partials + (size_t)KCHUNKS * B_ * N1;     // 8192 floats
    float* theta = h1 + B_ * N1;                             // 1536 floats

    // A: 4096 waves = 16 m-tiles x 256 K-chunks; 8 waves/block -> 512 blocks
    stn_gemm1_wmma<<<512, 256, 0, stream>>>(x, w1, partials);
    // B1: 8192 output elems
    stn_reduce_relu<<<(B_ * N1) / 256, 256, 0, stream>>>(partials, b1v, h1);
    // B2: one block, thread per batch row
    stn_mlp_tail<<<1, 256, 0, stream>>>(h1, w2, b2v, w3, b3v, theta);
    // C: 33.5M output elems
    stn_sample<<<(B_ * C_ * H_ * W_) / 256, 256, 0, stream>>>(x, theta, out);
}